// PairGeneration1_50345606644227
// MI455X (gfx1250) — compile-verified
//
#include <hip/hip_runtime.h>

// ---------------------------------------------------------------------------
// PairGeneration: out[b,i,j,o] = (text[b,i,:]·w2[o,:]) + (text[b,j,:]·w1[o,:]) + bias[o]
//   B=4, L=256, D=768, OUT=256.
// Phase 1: WMMA f32 GEMM -> P[2][1024][256] in d_ws (pj gets bias folded in).
// Phase 2: memory-bound broadcast add, 256 MB NT-streamed to HBM.
// ---------------------------------------------------------------------------

typedef __attribute__((ext_vector_type(2))) float v2f;
typedef __attribute__((ext_vector_type(4))) float v4f;
typedef __attribute__((ext_vector_type(8))) float v8f;

#define D_DIM    768      // K
#define M_DIM    1024     // B*L
#define O_DIM    256      // OUT (N)
#define W_STRIDE 1536     // weight row stride (2*D)

// ---------------------------------------------------------------------------
// Kernel 1: P[q][m][o] = sum_d text[m][d] * weight[o][q*768 + d]   (q=0 adds bias)
// One wave32 computes one 16x16 tile via V_WMMA_F32_16X16X4_F32.
//   A (16x4 f32):  lane holds M = lane&15, K = (lane>>4)*2 + v  (v = vgpr 0..1)
//   B (4x16 f32):  lane holds N = lane&15, K = (lane>>4)*2 + v  (w is OxK row-major,
//                  i.e. already the K x N operand column-per-lane)
//   C/D (16x16):   lane holds N = lane&15, M = v + (lane>>4)*8   (v = vgpr 0..7)
// 2048 tiles total (2 q * 64 mt * 16 ot), 4 waves per 128-thread block.
// ---------------------------------------------------------------------------
__global__ __launch_bounds__(128) void pairgen_wmma_gemm(
    const float* __restrict__ text,    // [1024][768]
    const float* __restrict__ weight,  // [256][1536]
    const float* __restrict__ bias,    // [256]
    float* __restrict__ P)             // [2][1024][256]
{
  const int lane = threadIdx.x & 31;
  const int wave = threadIdx.x >> 5;
  const int tile = blockIdx.x * 4 + wave;   // 0..2047
  const int q    = tile >> 10;              // 0: pj (w1, +bias), 1: pi (w2)
  const int mt   = (tile & 1023) >> 4;      // 0..63
  const int ot   = tile & 15;               // 0..15
  const int m0   = mt << 4;
  const int o0   = ot << 4;

  const int lm = lane & 15;                 // M for A / N for B / N for C
  const int kh = (lane >> 4) * 2;           // K sub-offset for this lane half

  const float* arow = text   + (size_t)(m0 + lm) * D_DIM    + kh;
  const float* brow = weight + (size_t)(o0 + lm) * W_STRIDE + (size_t)q * D_DIM + kh;

  v8f c = {};
  #pragma unroll 4
  for (int k = 0; k < D_DIM; k += 4) {
    v2f a = *(const v2f*)(arow + k);   // global_load_b64, L2-resident after 1st pass
    v2f b = *(const v2f*)(brow + k);
    // 8-arg form: (neg_a, A, neg_b, B, c_mod, C, reuse_a, reuse_b)
    c = __builtin_amdgcn_wmma_f32_16x16x4_f32(
        false, a, false, b, (short)0, c, false, false);
  }

  if (q == 0) {                             // fold bias into pj (depends on o == N only)
    const float bv = bias[o0 + lm];
    #pragma unroll
    for (int v = 0; v < 8; ++v) c[v] += bv;
  }

  // Scatter C: row m0 + v + (lane>>4)*8, col o0 + (lane&15). Lanes 0..15 cover
  // 16 consecutive floats per row -> 64B coalesced per half-wave. P is tiny (2MB).
  float* prow = P + (size_t)q * (M_DIM * O_DIM)
                  + (size_t)(m0 + ((lane >> 4) << 3)) * O_DIM + (o0 + lm);
  #pragma unroll
  for (int v = 0; v < 8; ++v) prow[(size_t)v * O_DIM] = c[v];
}

// ---------------------------------------------------------------------------
// Kernel 2: out[bi][j][o] = pi[bi][o] + pj[b*256 + j][o]      (bias already in pj)
// One 64-thread block per (b,i) row: pi row pinned in registers (1 float4/lane),
// pj rows stream from L2 (1 MB working set, reused 256x), output NT-streamed
// (256 MB > 192 MB L2 -> don't cache). Wave stores 32 contiguous float4 = 512B.
// ---------------------------------------------------------------------------
__global__ __launch_bounds__(64) void pairgen_bcast_add(
    const float* __restrict__ P,   // [2][1024][256]: P[0]=pj(+bias), P[1]=pi
    float* __restrict__ out)       // [1024][256][256]
{
  const int bi = blockIdx.x;       // b*256 + i, 0..1023
  const int o4 = threadIdx.x;      // float4 index along o, 0..63
  const int b  = bi >> 8;

  const v4f pi4 = ((const v4f*)(P + (size_t)M_DIM * O_DIM))[(size_t)bi * 64 + o4];
  const v4f* __restrict__ pj = (const v4f*)P + (size_t)(b << 8) * 64;
  v4f* __restrict__ orow = (v4f*)out + (size_t)bi * 256 * 64;

  #pragma unroll 4
  for (int j = 0; j < 256; ++j) {
    v4f r = pi4 + pj[j * 64 + o4];
    __builtin_nontemporal_store(r, &orow[j * 64 + o4]);  // global_store_b128 NT
  }
}

// ---------------------------------------------------------------------------
extern "C" void kernel_launch(void* const* d_in, const int* in_sizes, int n_in,
                              void* d_out, int out_size, void* d_ws, size_t ws_size,
                              hipStream_t stream) {
  const float* text   = (const float*)d_in[0];   // 4*256*768
  const float* weight = (const float*)d_in[1];   // 256*1536
  const float* bias   = (const float*)d_in[2];   // 256
  float*       out    = (float*)d_out;           // 4*256*256*256
  float*       P      = (float*)d_ws;            // needs 2*1024*256*4 = 2 MB

  (void)in_sizes; (void)n_in; (void)out_size; (void)ws_size;

  // Phase 1: 2048 wave-tiles, 4 waves (128 threads) per block -> 512 blocks.
  pairgen_wmma_gemm<<<512, 128, 0, stream>>>(text, weight, bias, P);
  // Phase 2: one block per (b,i) row.
  pairgen_bcast_add<<<1024, 64, 0, stream>>>(P, out);
}